// neural_network_equivariant_simple_62294205662140
// MI455X (gfx1250) — compile-verified
//
#include <hip/hip_runtime.h>

// ---------------------------------------------------------------------------
// Problem constants (from the reference)
// ---------------------------------------------------------------------------
#define EPSV     1e-6f
#define BN_EPSV  1e-5f
#define SLOPEV   0.2f
constexpr int Bn    = 2;
constexpr int Nn    = 4096;
constexpr int Kn    = 5;
constexpr int DLOW  = 6;    // N_VEC_LOW
constexpr int DH    = 40;   // N_VEC_HIGH
constexpr int FH    = 20;   // F_HIGH
constexpr int NL    = 2;    // NLAYERS

constexpr int KNN_WAVES = 4;               // waves per 16-row group
constexpr int COLS_PER_WAVE = Nn / KNN_WAVES;

typedef __attribute__((ext_vector_type(2))) float v2f;
typedef __attribute__((ext_vector_type(8))) float v8f;

// ---------------------------------------------------------------------------
// lift: y = xv @ W_lin^T  (per node: 6 -> 40), plus xx[n] = sum(y^2)
// ---------------------------------------------------------------------------
__global__ void lift_kernel(const float* __restrict__ x,
                            const float* __restrict__ Wlin,   // (40,6) row-major
                            float* __restrict__ Y,            // (B*N,40)
                            float* __restrict__ XX)           // (B*N)
{
    int node = blockIdx.x * blockDim.x + threadIdx.x;
    if (node >= Bn * Nn) return;
    float xv[DLOW];
#pragma unroll
    for (int j = 0; j < DLOW; ++j) xv[j] = x[node * DLOW + j];
    float ss = 0.f;
    for (int d = 0; d < DH; ++d) {
        float a = 0.f;
#pragma unroll
        for (int j = 0; j < DLOW; ++j) a += xv[j] * Wlin[d * DLOW + j];
        Y[(size_t)node * DH + d] = a;
        ss += a * a;
    }
    XX[node] = ss;
}

// ---------------------------------------------------------------------------
// branchless sorted top-5 insertion (descending)
// ---------------------------------------------------------------------------
__device__ __forceinline__ void ins5(float v, int j, float t[5], int ti[5]) {
#pragma unroll
    for (int p = 0; p < 5; ++p) {
        bool c = v > t[p];
        float tv = t[p]; int tj = ti[p];
        t[p]  = c ? v : tv;   ti[p] = c ? j : tj;
        v     = c ? tv : v;   j     = c ? tj : j;
    }
}

// ---------------------------------------------------------------------------
// load the 10 B fragments (16x4 f32 each) for column tile n0
// ---------------------------------------------------------------------------
__device__ __forceinline__ void loadB(const float* __restrict__ Yb, int n0,
                                      int mm, int half, v2f* bf)
{
#pragma unroll
    for (int kg = 0; kg < 10; ++kg)
        bf[kg] = *(const v2f*)(Yb + (size_t)(n0 + mm) * DH + kg * 4 + 2 * half);
}

// ---------------------------------------------------------------------------
// 10 chained WMMAs for one 16x16 tile + transpose via LDS + top-5 update
// ---------------------------------------------------------------------------
__device__ __forceinline__ void processTile(const v2f* afr, const v2f* bf,
                                            const float* __restrict__ XXb,
                                            int t0, int wave, int mm, int half,
                                            float tv[5], int ti[5],
                                            float (*tile)[16][17])
{
    v8f acc = {};
#pragma unroll
    for (int kg = 0; kg < 10; ++kg)
        acc = __builtin_amdgcn_wmma_f32_16x16x4_f32(
            false, afr[kg], false, bf[kg], (short)0, acc, false, false);

    // C layout: vgpr v, lane l -> M = v + 8*(l>=16), N = l&15
    const float xxn = XXb[t0 + mm];
#pragma unroll
    for (int v = 0; v < 8; ++v)
        tile[wave][v + 8 * half][mm] = 2.f * acc[v] - xxn;

    asm volatile("s_wait_dscnt 0x0" ::: "memory");

    const int cb = 8 * half;   // lane owns row mm; half-waves split columns
#pragma unroll
    for (int j = 0; j < 8; ++j)
        ins5(tile[wave][mm][cb + j], t0 + cb + j, tv, ti);

    asm volatile("s_wait_dscnt 0x0" ::: "memory");
}

// ---------------------------------------------------------------------------
// knn: per-batch Gram via V_WMMA_F32_16X16X4_F32, fused row-wise top-5.
// One block (4 waves) owns a 16-row group; each wave scans 1024 columns with
// double-buffered fragment loads; two-stage top-5 merge (half-lane, cross-wave).
// s = 2*inner(m,n) - xx[n]   (xx[m] is a per-row constant -> dropped)
// ---------------------------------------------------------------------------
__global__ void __launch_bounds__(128)
knn_kernel(const float* __restrict__ Y,    // (B*N,40)
           const float* __restrict__ XX,   // (B*N)
           int* __restrict__ IDX)          // (B*N,5) within-batch indices
{
    __shared__ float tile[KNN_WAVES][16][17];   // per-wave transpose buffer
    __shared__ float mv[KNN_WAVES][16][5];
    __shared__ int   mi[KNN_WAVES][16][5];

    const int wave = threadIdx.x >> 5;
    const int lane = threadIdx.x & 31;
    const int rgG  = blockIdx.x;               // [0, B*N/16)
    const int b    = rgG >> 8;                 // 256 row-groups per batch
    const int rg   = rgG & 255;
    const int m0   = rg << 4;

    const float* Yb  = Y  + (size_t)b * Nn * DH;
    const float* XXb = XX + (size_t)b * Nn;

    const int half = lane >> 4;
    const int mm   = lane & 15;

    // A fragments for the 16 rows (kept in registers for the whole sweep)
    v2f afr[10];
#pragma unroll
    for (int kg = 0; kg < 10; ++kg)
        afr[kg] = *(const v2f*)(Yb + (size_t)(m0 + mm) * DH + kg * 4 + 2 * half);

    float tvv[5]; int tii[5];
#pragma unroll
    for (int p = 0; p < 5; ++p) { tvv[p] = -3.0e38f; tii[p] = 0; }

    const int c0 = wave * COLS_PER_WAVE;
    const int c1 = c0 + COLS_PER_WAVE;

    // software-pipelined sweep: next tile's 10 loads issued before this tile's
    // WMMA + top-5 epilogue, so the epilogue hides the load latency
    v2f bf0[10], bf1[10];
    loadB(Yb, c0, mm, half, bf0);
    for (int t0 = c0; t0 < c1; t0 += 32) {
        loadB(Yb, t0 + 16, mm, half, bf1);
        if (t0 + 32 < c1)
            __builtin_prefetch(Yb + (size_t)(t0 + 32 + mm) * DH, 0, 3);
        processTile(afr, bf0, XXb, t0, wave, mm, half, tvv, tii, tile);
        if (t0 + 32 < c1)
            loadB(Yb, t0 + 32, mm, half, bf0);
        processTile(afr, bf1, XXb, t0 + 16, wave, mm, half, tvv, tii, tile);
    }

    // stage 1: merge the two half-wave lists for each row (intra-wave)
    if (half == 1) {
#pragma unroll
        for (int p = 0; p < 5; ++p) { mv[wave][mm][p] = tvv[p]; mi[wave][mm][p] = tii[p]; }
    }
    asm volatile("s_wait_dscnt 0x0" ::: "memory");
    if (half == 0) {
#pragma unroll
        for (int p = 0; p < 5; ++p) ins5(mv[wave][mm][p], mi[wave][mm][p], tvv, tii);
    }
    // stage 2: waves 1..3 publish their merged lists, wave 0 folds them in.
    // (wave w+1 candidates have strictly larger column indices; strict '>'
    //  insertion preserves lowest-index-first tie order)
    if (wave > 0 && half == 0) {
#pragma unroll
        for (int p = 0; p < 5; ++p) { mv[wave][mm][p] = tvv[p]; mi[wave][mm][p] = tii[p]; }
    }
    __syncthreads();
    if (wave == 0 && half == 0) {
#pragma unroll
        for (int w = 1; w < KNN_WAVES; ++w)
#pragma unroll
            for (int p = 0; p < 5; ++p) ins5(mv[w][mm][p], mi[w][mm][p], tvv, tii);
#pragma unroll
        for (int p = 0; p < 5; ++p)
            IDX[((size_t)b * Nn + m0 + mm) * Kn + p] = tii[p];
    }
}

// ---------------------------------------------------------------------------
// tiny zero kernel for the BN stat accumulators
// ---------------------------------------------------------------------------
__global__ void zero_kernel(float* p, int n) {
    int i = blockIdx.x * blockDim.x + threadIdx.x;
    if (i < n) p[i] = 0.f;
}

// ---------------------------------------------------------------------------
// stage center row + 5 gathered neighbor rows into LDS (blockDim = 256)
// ---------------------------------------------------------------------------
__device__ __forceinline__ void stage_node(const float* __restrict__ Y,
                                           const int* __restrict__ IDX,
                                           int b, int n,
                                           float* sc, float* snb /* [5*40] */)
{
    const int tid = threadIdx.x;
    if (tid < DH) sc[tid] = Y[((size_t)b * Nn + n) * DH + tid];
    if (tid >= DH && tid < DH + Kn * DH) {
        int k = (tid - DH) / DH, d = (tid - DH) % DH;
        int nb = IDX[((size_t)b * Nn + n) * Kn + k];
        snb[k * DH + d] = Y[((size_t)b * Nn + nb) * DH + d];
    }
}

// ---------------------------------------------------------------------------
// VN block 1 BN statistics: sum / sumsq of ||p||_d per output channel (20)
// ---------------------------------------------------------------------------
__global__ void __launch_bounds__(256)
stats1_kernel(const float* __restrict__ Y, const int* __restrict__ IDX,
              const float* __restrict__ W1, float* __restrict__ stats)
{
    __shared__ float sc[DH], snb[Kn * DH];
    __shared__ float s1[FH], s2[FH];
    const int node = blockIdx.x;
    const int b = node / Nn, n = node % Nn;
    const int tid = threadIdx.x;

    stage_node(Y, IDX, b, n, sc, snb);
    if (tid < FH) { s1[tid] = 0.f; s2[tid] = 0.f; }
    __syncthreads();

    if (tid < FH * Kn) {
        int o = tid / Kn, k = tid % Kn;
        float w0 = W1[o * 2 + 0], w1 = W1[o * 2 + 1];
        float sp2 = 0.f;
        for (int d = 0; d < DH; ++d) {
            float c = sc[d], e = snb[k * DH + d] - c;
            float p = w0 * e + w1 * c;
            sp2 += p * p;
        }
        float nrm = sqrtf(sp2) + EPSV;
        atomicAdd(&s1[o], nrm);
        atomicAdd(&s2[o], nrm * nrm);
    }
    __syncthreads();
    if (tid < FH) {
        atomicAdd(&stats[tid], s1[tid]);
        atomicAdd(&stats[FH + tid], s2[tid]);
    }
}

// ---------------------------------------------------------------------------
// compute f1[(o*5+k)*40+d] into LDS given staged inputs + finalized BN1 stats
// ---------------------------------------------------------------------------
__device__ __forceinline__ void compute_f1(const float* sc, const float* snb,
                                           const float* __restrict__ W1,
                                           const float* __restrict__ D1,
                                           const float* __restrict__ g1,
                                           const float* __restrict__ b1,
                                           const float* __restrict__ stats,
                                           float* f1)
{
    const int tid = threadIdx.x;
    if (tid < FH * Kn) {
        int o = tid / Kn, k = tid % Kn;
        const float cnt = (float)(Bn * Nn * Kn);
        float mu  = stats[o] / cnt;
        float var = stats[FH + o] / cnt - mu * mu;
        float iv  = rsqrtf(var + BN_EPSV);
        float w0 = W1[o * 2 + 0], w1 = W1[o * 2 + 1];
        float dw0 = D1[o * 2 + 0], dw1 = D1[o * 2 + 1];
        float sp2 = 0.f, spd = 0.f, sd2 = 0.f;
        for (int d = 0; d < DH; ++d) {
            float c = sc[d], e = snb[k * DH + d] - c;
            float p  = w0 * e + w1 * c;
            float dv = dw0 * e + dw1 * c;
            sp2 += p * p; spd += p * dv; sd2 += dv * dv;
        }
        float nrm   = sqrtf(sp2) + EPSV;
        float nbn   = (nrm - mu) * iv * g1[o] + b1[o];
        float scale = nbn / nrm;
        float dot   = scale * spd;
        float coef  = (dot < 0.f) ? (1.f - SLOPEV) * dot / (sd2 + EPSV) : 0.f;
        for (int d = 0; d < DH; ++d) {
            float c = sc[d], e = snb[k * DH + d] - c;
            float p  = w0 * e + w1 * c;
            float dv = dw0 * e + dw1 * c;
            f1[(o * Kn + k) * DH + d] = scale * p - coef * dv;
        }
    }
}

// ---------------------------------------------------------------------------
// VN block 2 BN statistics (2 channels)
// ---------------------------------------------------------------------------
__global__ void __launch_bounds__(256)
stats2_kernel(const float* __restrict__ Y, const int* __restrict__ IDX,
              const float* __restrict__ W1, const float* __restrict__ D1,
              const float* __restrict__ g1, const float* __restrict__ b1,
              const float* __restrict__ stats1, const float* __restrict__ W2,
              float* __restrict__ stats2)
{
    __shared__ float sc[DH], snb[Kn * DH];
    __shared__ float f1[FH * Kn * DH];
    __shared__ float s1[2], s2[2];
    const int node = blockIdx.x;
    const int b = node / Nn, n = node % Nn;
    const int tid = threadIdx.x;

    stage_node(Y, IDX, b, n, sc, snb);
    if (tid < 2) { s1[tid] = 0.f; s2[tid] = 0.f; }
    __syncthreads();
    compute_f1(sc, snb, W1, D1, g1, b1, stats1, f1);
    __syncthreads();

    if (tid < 2 * Kn) {
        int o = tid / Kn, k = tid % Kn;
        float sp2 = 0.f;
        for (int d = 0; d < DH; ++d) {
            float p2 = 0.f;
#pragma unroll
            for (int f = 0; f < FH; ++f)
                p2 += W2[o * FH + f] * f1[(f * Kn + k) * DH + d];
            sp2 += p2 * p2;
        }
        float nrm = sqrtf(sp2) + EPSV;
        atomicAdd(&s1[o], nrm);
        atomicAdd(&s2[o], nrm * nrm);
    }
    __syncthreads();
    if (tid < 2) {
        atomicAdd(&stats2[tid], s1[tid]);
        atomicAdd(&stats2[2 + tid], s2[tid]);
    }
}

// ---------------------------------------------------------------------------
// apply: f1 -> block2 -> mean over k -> Wv contraction -> residual update.
// Writes Ynext + its xx; on the last layer also projects back to 6 dims.
// ---------------------------------------------------------------------------
__global__ void __launch_bounds__(256)
apply_kernel(const float* __restrict__ Yin, const int* __restrict__ IDX,
             const float* __restrict__ W1, const float* __restrict__ D1,
             const float* __restrict__ g1, const float* __restrict__ b1,
             const float* __restrict__ stats1,
             const float* __restrict__ W2, const float* __restrict__ D2,
             const float* __restrict__ g2, const float* __restrict__ b2,
             const float* __restrict__ stats2,
             const float* __restrict__ Wv, const float* __restrict__ hbuf,
             int layer,
             float* __restrict__ Yout, float* __restrict__ XXout,
             const float* __restrict__ Wlin, float* __restrict__ out,
             int writeOut)
{
    __shared__ float sc[DH], snb[Kn * DH];
    __shared__ float f1[FH * Kn * DH];
    __shared__ float scl2[2][Kn], cf2[2][Kn];
    __shared__ float yrow[DH];
    __shared__ float sxx;
    const int node = blockIdx.x;
    const int b = node / Nn, n = node % Nn;
    const int tid = threadIdx.x;

    stage_node(Yin, IDX, b, n, sc, snb);
    if (tid == 0) sxx = 0.f;
    __syncthreads();
    compute_f1(sc, snb, W1, D1, g1, b1, stats1, f1);
    __syncthreads();

    if (tid < 2 * Kn) {
        int o = tid / Kn, k = tid % Kn;
        const float cnt = (float)(Bn * Nn * Kn);
        float mu  = stats2[o] / cnt;
        float var = stats2[2 + o] / cnt - mu * mu;
        float iv  = rsqrtf(var + BN_EPSV);
        float sp2 = 0.f, spd = 0.f, sd2 = 0.f;
        for (int d = 0; d < DH; ++d) {
            float p2 = 0.f, q2 = 0.f;
#pragma unroll
            for (int f = 0; f < FH; ++f) {
                float v = f1[(f * Kn + k) * DH + d];
                p2 += W2[o * FH + f] * v;
                q2 += D2[o * FH + f] * v;
            }
            sp2 += p2 * p2; spd += p2 * q2; sd2 += q2 * q2;
        }
        float nrm   = sqrtf(sp2) + EPSV;
        float nbn   = (nrm - mu) * iv * g2[o] + b2[o];
        float scale = nbn / nrm;
        float dot   = scale * spd;
        float coef  = (dot < 0.f) ? (1.f - SLOPEV) * dot / (sd2 + EPSV) : 0.f;
        scl2[o][k] = scale;
        cf2[o][k]  = coef;
    }
    __syncthreads();

    if (tid < DH) {
        int d = tid;
        float accO[2] = {0.f, 0.f};
#pragma unroll
        for (int o = 0; o < 2; ++o) {
#pragma unroll
            for (int k = 0; k < Kn; ++k) {
                float p2 = 0.f, q2 = 0.f;
#pragma unroll
                for (int f = 0; f < FH; ++f) {
                    float v = f1[(f * Kn + k) * DH + d];
                    p2 += W2[o * FH + f] * v;
                    q2 += D2[o * FH + f] * v;
                }
                accO[o] += scl2[o][k] * p2 - cf2[o][k] * q2;
            }
        }
        float y1_0 = accO[0] * (1.f / (float)Kn);
        float y1_1 = accO[1] * (1.f / (float)Kn);
        float ynew = Wv[0] * y1_0 + Wv[1] * y1_1;
        float h = hbuf[layer];
        float yo = sc[d] + h * h * ynew;
        Yout[((size_t)b * Nn + n) * DH + d] = yo;
        yrow[d] = yo;
        atomicAdd(&sxx, yo * yo);
    }
    __syncthreads();
    if (tid == 0) XXout[(size_t)b * Nn + n] = sxx;
    if (writeOut && tid < DLOW) {
        float a = 0.f;
        for (int d = 0; d < DH; ++d) a += yrow[d] * Wlin[d * DLOW + tid];
        out[((size_t)b * Nn + n) * DLOW + tid] = a;
    }
}

// ---------------------------------------------------------------------------
// launcher
// ---------------------------------------------------------------------------
extern "C" void kernel_launch(void* const* d_in, const int* in_sizes, int n_in,
                              void* d_out, int out_size, void* d_ws, size_t ws_size,
                              hipStream_t stream)
{
    const float* x    = (const float*)d_in[0];
    // d_in[1] node_attr unused
    const float* Wlin = (const float*)d_in[2];
    const float* h    = (const float*)d_in[3];
    const float* W1   = (const float*)d_in[4];
    const float* D1   = (const float*)d_in[5];
    const float* g1   = (const float*)d_in[6];
    const float* b1   = (const float*)d_in[7];
    const float* W2   = (const float*)d_in[8];
    const float* D2   = (const float*)d_in[9];
    const float* g2   = (const float*)d_in[10];
    const float* b2   = (const float*)d_in[11];
    const float* Wv   = (const float*)d_in[12];
    // batch / edge_src / edge_dst unused
    float* out = (float*)d_out;

    float* ws = (float*)d_ws;
    float* YA     = ws;                         // B*N*40
    float* YB     = YA + (size_t)Bn * Nn * DH;  // B*N*40
    float* XX     = YB + (size_t)Bn * Nn * DH;  // B*N
    float* stats1 = XX + (size_t)Bn * Nn;       // 40 (sum,sumsq)
    float* stats2 = stats1 + 2 * FH;            // 4
    int*   IDX    = (int*)(stats2 + 8);         // B*N*5

    lift_kernel<<<(Bn * Nn + 255) / 256, 256, 0, stream>>>(x, Wlin, YA, XX);

    float* Ycur = YA;
    float* Ynxt = YB;
    for (int layer = 0; layer < NL; ++layer) {
        knn_kernel<<<Bn * Nn / 16, 128, 0, stream>>>(Ycur, XX, IDX);
        zero_kernel<<<1, 64, 0, stream>>>(stats1, 2 * FH + 4);
        stats1_kernel<<<Bn * Nn, 256, 0, stream>>>(
            Ycur, IDX, W1 + layer * FH * 2, stats1);
        stats2_kernel<<<Bn * Nn, 256, 0, stream>>>(
            Ycur, IDX, W1 + layer * FH * 2, D1 + layer * FH * 2,
            g1 + layer * FH, b1 + layer * FH, stats1,
            W2 + layer * 2 * FH, stats2);
        apply_kernel<<<Bn * Nn, 256, 0, stream>>>(
            Ycur, IDX, W1 + layer * FH * 2, D1 + layer * FH * 2,
            g1 + layer * FH, b1 + layer * FH, stats1,
            W2 + layer * 2 * FH, D2 + layer * 2 * FH,
            g2 + layer * 2, b2 + layer * 2, stats2,
            Wv + layer * 2, h, layer,
            Ynxt, XX, Wlin, out, (layer == NL - 1) ? 1 : 0);
        float* t = Ycur; Ycur = Ynxt; Ynxt = t;
    }
}